// SAGPool_50130858279307
// MI455X (gfx1250) — compile-verified
//
#include <hip/hip_runtime.h>
#include <hip/hip_bf16.h>
#include <math.h>

// ---------------- problem constants (from reference) ----------------
#define NV 2000000      // vertices
#define NE 6000000      // edges
#define NC 128          // channels
#define NB 64           // meshes

typedef __attribute__((ext_vector_type(2))) float    v2f;
typedef __attribute__((ext_vector_type(8))) float    v8f;
typedef __attribute__((ext_vector_type(4))) unsigned u32x4;
typedef __attribute__((ext_vector_type(8))) unsigned u32x8;

#define RSTRIDE 132            // LDS row stride in floats (128 + 4 pad, via TDM pad feature)
#define WAVES_PER_BLOCK 2
#define LINEAR_BLOCKS 2048     // 4096 waves -> ~30 tiles each (double-buffered TDM pays off)
#define SEG_INIT 0x007FFFFFu   // monotone-mapped -inf

// monotone map: float bits -> orderable u32 (max over u32 == max over float)
__device__ __forceinline__ unsigned fmap(float f) {
    unsigned b = __float_as_uint(f);
    return (b & 0x80000000u) ? ~b : (b | 0x80000000u);
}
__device__ __forceinline__ float funmap(unsigned k) {
    unsigned b = (k & 0x80000000u) ? (k & 0x7FFFFFFFu) : ~k;
    return __uint_as_float(b);
}

__device__ __forceinline__ unsigned rfl(unsigned x) {
    return (unsigned)__builtin_amdgcn_readfirstlane((int)x);
}

// ---------------- TDM: async copy of one 16x128-f32 tile (row-padded) into LDS ----------------
// D# per CDNA5 ISA: group0 = {count|..., lds_addr, global_addr[31:0], global_addr[56:32]|type=2}
// group1: data_size=4B, pad_enable, pad_interval=6 (128 DW), pad_amount=3 (4 DW) -> 132-float rows,
//         tensor_dim0=128, tensor_dim1=NV, tile_dim0=128, tile_dim1=16, tensor_dim0_stride=128.
__device__ __forceinline__ void tdm_load_tile(const float* gbase, unsigned ldsOff) {
    unsigned long long ga = (unsigned long long)(uintptr_t)gbase;
    u32x4 g0 = { 1u,                                     // count=1, no gather
                 rfl(ldsOff),                            // lds_addr (bytes)
                 rfl((unsigned)ga),                      // global_addr[31:0]
                 rfl((unsigned)(ga >> 32) & 0x01FFFFFFu) | (2u << 30) }; // addr[56:32] | type=2
    u32x8 g1 = { (2u << 16) | (1u << 20) | (6u << 22) | (3u << 25), // data_size=4B, pad 4DW/128DW
                 128u << 16,                             // tensor_dim0[15:0]=128 (no atomic barrier)
                 ((unsigned)NV & 0xFFFFu) << 16,         // tensor_dim0 hi=0 | tensor_dim1 lo
                 ((unsigned)NV >> 16) | (128u << 16),    // tensor_dim1 hi | tile_dim0=128
                 16u,                                    // tile_dim1=16, tile_dim2=0
                 128u,                                   // tensor_dim0_stride=128 elements
                 0u, 0u };
    u32x4 gz = { 0u, 0u, 0u, 0u };                       // groups 2/3: dims unused (2D)
    asm volatile("tensor_load_to_lds %0, %1, %2, %3"
                 :: "s"(g0), "s"(g1), "s"(gz), "s"(gz)
                 : "memory");
}

// ---------------- kernel 0: init workspace ----------------
__global__ void k_init(float* __restrict__ nbr, unsigned* __restrict__ segkeys) {
    int i = blockIdx.x * blockDim.x + threadIdx.x;
    if (i < NV) nbr[i] = 0.0f;
    if (i < NB) segkeys[i] = SEG_INIT;
}

// ---------------- kernel 1: h0/h1 = verts @ [w0;w1]^T + b via WMMA f32 16x16x4 ----------------
// A (16x4) = weight rows (M=0 -> w0, M=1 -> w1, rest 0), preloaded in registers (loop-invariant)
// B (4x16) = 16-vertex tile chunk, streamed global->LDS by the Tensor Data Mover (double-buffered)
// D row 0 = h0, row 1 = h1 for the 16 vertices -> acc[0]/acc[1] of lanes 0..15 (coalesced store)
__global__ void __launch_bounds__(WAVES_PER_BLOCK * 32)
k_linear_wmma(const float* __restrict__ verts,
              const float* __restrict__ w0, const float* __restrict__ b0,
              const float* __restrict__ w1, const float* __restrict__ b1,
              float* __restrict__ h0, float* __restrict__ h1, int ntiles) {
    __shared__ __attribute__((aligned(16))) float vbuf[WAVES_PER_BLOCK][2][16 * RSTRIDE];

    const int lane = threadIdx.x & 31;
    const int wv   = threadIdx.x >> 5;
    const int m    = lane & 15;       // A-row / B-column index for this lane
    const int half = lane >> 4;       // K sub-pair selector (0 -> k0,k1 ; 1 -> k2,k3)
    const float bias0 = b0[0];
    const float bias1 = b1[0];

    // preload invariant A fragments: lane pattern (m, half) -> w_m[4kc + 2*half .. +1]
    v2f afrag[32];
    const float* wrow = (m == 0) ? w0 : (m == 1) ? w1 : (const float*)0;
    #pragma unroll
    for (int kc = 0; kc < 32; ++kc) {
        v2f a = {0.0f, 0.0f};
        if (wrow) a = *(const v2f*)(wrow + 4 * kc + 2 * half);
        afrag[kc] = a;
    }

    // LDS byte offsets of the two per-wave tile buffers (low 32 bits of generic LDS address)
    unsigned ldsOff[2];
    ldsOff[0] = (unsigned)(uintptr_t)(void*)&vbuf[wv][0][0];
    ldsOff[1] = (unsigned)(uintptr_t)(void*)&vbuf[wv][1][0];

    const int S = (gridDim.x * blockDim.x) >> 5;                 // total waves
    int t = (blockIdx.x * blockDim.x + threadIdx.x) >> 5;        // wave id -> first tile
    if (t >= ntiles) return;

    tdm_load_tile(verts + (size_t)t * 16 * NC, ldsOff[0]);       // prologue
    int p = 0;
    while (true) {
        const int tn = t + S;
        const bool more = tn < ntiles;
        if (more) tdm_load_tile(verts + (size_t)tn * 16 * NC, ldsOff[p ^ 1]);
        // wait for tile t's TDM (in-order per wave); asm+memory keeps ds_loads below the wait
        if (more) asm volatile("s_wait_tensorcnt 0x1" ::: "memory");
        else      asm volatile("s_wait_tensorcnt 0x0" ::: "memory");

        const float* bP = &vbuf[wv][p][0] + m * RSTRIDE + 2 * half;
        v8f acc = {};
        #pragma unroll
        for (int kc = 0; kc < 32; ++kc) {
            v2f b = *(const v2f*)(bP + 4 * kc);   // B[4kc+2*half .. +1][n=m]
            acc = __builtin_amdgcn_wmma_f32_16x16x4_f32(
                false, afrag[kc], false, b, (short)0, acc, false, false);
        }

        const long vbase = (long)t * 16;
        if (lane < 16) {                          // D row0 -> h0, row1 -> h1 (coalesced)
            h0[vbase + lane] = acc[0] + bias0;
            h1[vbase + lane] = acc[1] + bias1;
        }
        // all B-frag ds_loads must retire before next TDM overwrites this buffer
        asm volatile("s_wait_dscnt 0x0" ::: "memory");
        if (!more) break;
        t = tn;
        p ^= 1;
    }
}

// ---------------- kernel 2: edge scatter (h1 gathers + float atomic adds, L2-resident) ----------------
__global__ void k_edges(const long long* __restrict__ edges,
                        const float* __restrict__ h1, float* __restrict__ nbr) {
    int e = blockIdx.x * blockDim.x + threadIdx.x;
    if (e >= NE) return;
    int s = (int)edges[2 * e];
    int d = (int)edges[2 * e + 1];
    float hs = h1[s];
    float hd = h1[d];
    __hip_atomic_fetch_add(&nbr[s], hd, __ATOMIC_RELAXED, __HIP_MEMORY_SCOPE_AGENT);
    __hip_atomic_fetch_add(&nbr[d], hs, __ATOMIC_RELAXED, __HIP_MEMORY_SCOPE_AGENT);
}

// ---------------- kernel 3: scores = tanh(h0+nbr); hierarchical segment max ----------------
__global__ void __launch_bounds__(1024)
k_score_segmax(const float* __restrict__ h0, const float* __restrict__ nbr,
               const long long* __restrict__ vidx, unsigned* __restrict__ segkeys) {
    __shared__ unsigned smax[NB];
    if (threadIdx.x < NB) smax[threadIdx.x] = SEG_INIT;
    __syncthreads();

    int v = blockIdx.x * blockDim.x + threadIdx.x;
    if (v < NV) {
        float sc = tanhf(h0[v] + nbr[v]);
        atomicMax(&smax[(int)vidx[v]], fmap(sc));
    }
    __syncthreads();

    // verts_idx is sorted -> each block touches ~1-2 meshes; skip untouched slots
    if (threadIdx.x < NB) {
        unsigned k = smax[threadIdx.x];
        if (k != SEG_INIT) atomicMax(&segkeys[threadIdx.x], k);
    }
}

// ---------------- kernel 4: unmap keys -> float output ----------------
__global__ void k_finish(const unsigned* __restrict__ segkeys, float* __restrict__ out) {
    int i = threadIdx.x;
    if (i < NB) out[i] = funmap(segkeys[i]);
}

extern "C" void kernel_launch(void* const* d_in, const int* in_sizes, int n_in,
                              void* d_out, int out_size, void* d_ws, size_t ws_size,
                              hipStream_t stream) {
    const float*     verts = (const float*)d_in[0];
    const long long* edges = (const long long*)d_in[1];
    const long long* vidx  = (const long long*)d_in[2];
    // d_in[3] = edges_idx (unused by reference), d_in[4] = num_meshes (== 64)
    const float* w0 = (const float*)d_in[5];
    const float* b0 = (const float*)d_in[6];
    const float* w1 = (const float*)d_in[7];
    const float* b1 = (const float*)d_in[8];
    float* out = (float*)d_out;

    float*    h0      = (float*)d_ws;
    float*    h1      = h0 + NV;
    float*    nbr     = h1 + NV;
    unsigned* segkeys = (unsigned*)(nbr + NV);

    // init
    k_init<<<(NV + 255) / 256, 256, 0, stream>>>(nbr, segkeys);

    // linear: TDM double-buffered, one wave -> many 16-vertex tiles
    const int ntiles = NV / 16;                               // 125000
    k_linear_wmma<<<LINEAR_BLOCKS, WAVES_PER_BLOCK * 32, 0, stream>>>(
        verts, w0, b0, w1, b1, h0, h1, ntiles);

    // edge scatter
    k_edges<<<(NE + 255) / 256, 256, 0, stream>>>(edges, h1, nbr);

    // scores + segment max
    k_score_segmax<<<(NV + 1023) / 1024, 1024, 0, stream>>>(h0, nbr, vidx, segkeys);

    // finish
    k_finish<<<1, 64, 0, stream>>>(segkeys, out);
}